// GraphConvolutionK_78950088835483
// MI455X (gfx1250) — compile-verified
//
#include <hip/hip_runtime.h>

typedef __attribute__((ext_vector_type(2))) float v2f;
typedef __attribute__((ext_vector_type(8))) float v8f;

// -----------------------------------------------------------------------------
// Dense GEMM: support[M x 64] = X[M x 256] * W[256 x 64]   (M = N*K = 200000)
// One wave32 computes a 16x64 tile via V_WMMA_F32_16X16X4_F32 (fp32-exact).
//
// W is staged in LDS pre-packed for the WMMA B-fragment layout:
//   row-pair p = K/2, column n  ->  wlds[p*160 + n*2 + (K&1)]
// so a B fragment (W[K][n], W[K+1][n]) is ONE aligned 8B LDS load straight
// into an even VGPR pair (no repack movs). Row-pair stride 160 floats = 640B
// = 32 banks (mod 64), so the even/odd row-pairs read by the two half-waves
// occupy disjoint bank halves -> minimal-conflict ds_load_2addr_b64.
// Padded W = 128*160*4 = 80KB -> dynamic LDS (gfx1250 allows 320KB/WG).
// -----------------------------------------------------------------------------
__global__ __launch_bounds__(256) void gcn_gemm(const float* __restrict__ x,
                                                const float* __restrict__ w,
                                                float* __restrict__ support,
                                                int mtiles) {
  extern __shared__ float wlds[];  // 128 row-pairs * 160 floats = 80KB
  {
    const int t = threadIdx.x;
#pragma unroll
    for (int i = 0; i < 32; ++i) {
      int q = t + i * 256;        // pair id: 0..8191
      int p = q >> 6;             // row-pair 0..127
      int n = q & 63;             // column
      v2f pr;
      pr.x = w[(2 * p) * 64 + n];      // W[2p][n]   (coalesced across lanes)
      pr.y = w[(2 * p + 1) * 64 + n];  // W[2p+1][n]
      *(v2f*)(wlds + p * 160 + n * 2) = pr;   // ds_store_b64
    }
  }
  __syncthreads();

  const int wave = blockIdx.x * 8 + (threadIdx.x >> 5);
  if (wave >= mtiles) return;
  const int lane = threadIdx.x & 31;
  const int half = lane >> 4;   // 0: lanes 0-15, 1: lanes 16-31
  const int koff = half << 1;   // A/B K sub-offset: 0 or 2
  const int l15  = lane & 15;

  // A fragment (32-bit A 16x4): lanes 0-15 hold K={d0,d0+1}, lanes 16-31
  // K={d0+2,d0+3} -> contiguous 8B per lane from row-major X.
  const float* xrow  = x + (size_t)(wave * 16 + l15) * 256 + koff;
  // Per-lane LDS base: row-pair (d0/2 + half), column l15.
  const float* wbase = wlds + (size_t)half * 160 + (size_t)l15 * 2;

  v8f c0 = {}, c1 = {}, c2 = {}, c3 = {};
#pragma unroll 4
  for (int d0 = 0; d0 < 256; d0 += 4) {
    v2f a = *(const v2f*)(xrow + d0);
    const float* wp = wbase + (d0 >> 1) * 160;
    v2f b0 = *(const v2f*)(wp + 0);    // cols  0..15 : pair slot l15
    v2f b1 = *(const v2f*)(wp + 32);   // cols 16..31 : pair slot l15+16
    v2f b2 = *(const v2f*)(wp + 64);   // cols 32..47 : pair slot l15+32
    v2f b3 = *(const v2f*)(wp + 96);   // cols 48..63 : pair slot l15+48
    c0 = __builtin_amdgcn_wmma_f32_16x16x4_f32(false, a, false, b0, (short)0, c0, false, false);
    c1 = __builtin_amdgcn_wmma_f32_16x16x4_f32(false, a, false, b1, (short)0, c1, false, false);
    c2 = __builtin_amdgcn_wmma_f32_16x16x4_f32(false, a, false, b2, (short)0, c2, false, false);
    c3 = __builtin_amdgcn_wmma_f32_16x16x4_f32(false, a, false, b3, (short)0, c3, false, false);
  }

  // C/D layout: VGPR j -> M = j + 8*half, N = l15 (+16 per tile)
  float* op = support + (size_t)wave * 16 * 64 + (size_t)(half << 3) * 64 + l15;
#pragma unroll
  for (int j = 0; j < 8; ++j) {
    op[j * 64 + 0]  = c0[j];
    op[j * 64 + 16] = c1[j];
    op[j * 64 + 32] = c2[j];
    op[j * 64 + 48] = c3[j];
  }
}

// -----------------------------------------------------------------------------
// Zero-fill d_out (harness poisons it; atomics need a zeroed accumulator).
// -----------------------------------------------------------------------------
__global__ __launch_bounds__(256) void gcn_zero(float4* __restrict__ out, int n4) {
  int i = blockIdx.x * 256 + threadIdx.x;
  if (i < n4) out[i] = make_float4(0.f, 0.f, 0.f, 0.f);
}

// -----------------------------------------------------------------------------
// Edge scatter: out[r,:] += vals[e] * support[c,:]  (256 floats per edge).
// 64 threads per edge, one float4 each; support & out are L2-resident
// (51.2MB each, global L2 = 192MB) so gathers + f32 atomics run at L2 speed.
// -----------------------------------------------------------------------------
__global__ __launch_bounds__(256) void gcn_scatter(const int* __restrict__ rows,
                                                   const int* __restrict__ cols,
                                                   const float* __restrict__ vals,
                                                   const float4* __restrict__ support4,
                                                   float* __restrict__ out, int E) {
  int e = blockIdx.x * 4 + (threadIdx.x >> 6);
  if (e >= E) return;
  int q = threadIdx.x & 63;
  int r = rows[e];
  int c = cols[e];
  float v = vals[e];
  float4 s = support4[(size_t)c * 64 + q];
  float* dst = out + (size_t)r * 256 + (size_t)q * 4;
  atomicAdd(dst + 0, v * s.x);
  atomicAdd(dst + 1, v * s.y);
  atomicAdd(dst + 2, v * s.z);
  atomicAdd(dst + 3, v * s.w);
}

// -----------------------------------------------------------------------------
// In-place ReLU over d_out.
// -----------------------------------------------------------------------------
__global__ __launch_bounds__(256) void gcn_relu(float4* __restrict__ out, int n4) {
  int i = blockIdx.x * 256 + threadIdx.x;
  if (i < n4) {
    float4 t = out[i];
    t.x = fmaxf(t.x, 0.f);
    t.y = fmaxf(t.y, 0.f);
    t.z = fmaxf(t.z, 0.f);
    t.w = fmaxf(t.w, 0.f);
    out[i] = t;
  }
}

extern "C" void kernel_launch(void* const* d_in, const int* in_sizes, int n_in,
                              void* d_out, int out_size, void* d_ws, size_t ws_size,
                              hipStream_t stream) {
  const float* x    = (const float*)d_in[0];  // [N, K, 256]
  const float* w    = (const float*)d_in[1];  // [256, 64]
  const int*   rows = (const int*)d_in[2];    // [E]
  const int*   cols = (const int*)d_in[3];    // [E]
  const float* vals = (const float*)d_in[4];  // [E]
  float* out     = (float*)d_out;             // [N, K, 64]
  float* support = (float*)d_ws;              // scratch: [N*K, 64] = 51.2MB

  const int M      = in_sizes[0] / 256;  // N*K = 200000
  const int mtiles = M / 16;             // 12500 (exact)
  const int E      = in_sizes[2];        // 800000
  const int n4     = out_size / 4;       // 3.2M float4

  const size_t wlds_bytes = 128 * 160 * sizeof(float);  // 80KB padded W

  gcn_zero<<<(n4 + 255) / 256, 256, 0, stream>>>((float4*)out, n4);
  gcn_gemm<<<(mtiles + 7) / 8, 256, wlds_bytes, stream>>>(x, w, support, mtiles);
  gcn_scatter<<<(E + 3) / 4, 256, 0, stream>>>(rows, cols, vals,
                                               (const float4*)support, out, E);
  gcn_relu<<<(n4 + 255) / 256, 256, 0, stream>>>((float4*)out, n4);
}